// GAT_3393024164342
// MI455X (gfx1250) — compile-verified
//
#include <hip/hip_runtime.h>
#include <math.h>

// ---------------- problem constants (match reference) ----------------
#define NN   8192      // nodes
#define EE   131072    // edges
#define FIN  512
#define HH   256
#define CCL  16        // clusters / final dim
#define NEGS 0.2f
#define BNEPS 1e-5f

typedef __attribute__((ext_vector_type(2))) float v2f;
typedef __attribute__((ext_vector_type(8))) float v8f;

// monotone float <-> orderable u32 (for atomicMax-based segment max)
__device__ __forceinline__ unsigned f2key(float f) {
  unsigned u = __float_as_uint(f);
  return (u & 0x80000000u) ? ~u : (u | 0x80000000u);
}
__device__ __forceinline__ float key2f(unsigned k) {
  unsigned u = (k & 0x80000000u) ? (k & 0x7FFFFFFFu) : ~k;
  return __uint_as_float(u);
}

// ---------------------------------------------------------------------
// WMMA f32 GEMM: C[M x Nc] = A[M x K] (row-major) @ B[K x Nc] (row-major)
// Block = 128 threads = 4 waves sharing one 16-column B panel staged in LDS
// (K*16 floats <= 32KB, well under the 320KB/WGP LDS). Wave w computes the
// 16x16 tile at row (blockIdx.x*4+w)*16, col blockIdx.y*16 with
// V_WMMA_F32_16X16X4_F32 along K.
// A frag: lane(0..15) = row M, VGPR pair = K {0,1}; lane(16..31) = K {2,3}.
// B frag: lane&15 = col N, lane half selects K pair (mirrors A).
// D: vgpr r -> row (half*8 + r), col = lane&15.
// ---------------------------------------------------------------------
__global__ void gat_wmma_gemm_f32(const float* __restrict__ A,
                                  const float* __restrict__ B,
                                  const float* __restrict__ bias,
                                  float* __restrict__ C,
                                  int M, int K, int Nc) {
  __shared__ float Bs[FIN * 16];                    // max K = 512 -> 32KB

  const int tid  = threadIdx.x;
  const int lane = tid & 31;
  const int wv   = tid >> 5;                        // 0..3
  const int tm   = (blockIdx.x * 4 + wv) << 4;
  const int tn   = blockIdx.y << 4;

  // cooperative B-panel stage: K rows x 16 cols, float4 granularity.
  // row k of the panel is 16 contiguous, 64B-aligned floats in B.
  {
    const int nf4 = K * 4;                          // float4 units in panel
    for (int u = tid; u < nf4; u += 128) {
      int k  = u >> 2;
      int c4 = (u & 3) << 2;
      *(float4*)&Bs[k * 16 + c4] =
          *(const float4*)&B[(size_t)k * Nc + tn + c4];
    }
  }
  __syncthreads();

  const int half = lane >> 4;                       // 0 | 1
  const int l15  = lane & 15;
  const int koff = half * 2;

  const float* __restrict__ Arow = A + (size_t)(tm + l15) * K;

  v8f acc = {};
  for (int k = 0; k < K; k += 4) {
    v2f a, b;
    a.x = Arow[k + koff];                           // aligned 8B pair
    a.y = Arow[k + koff + 1];
    b.x = Bs[(k + koff) * 16 + l15];                // conflict-free ds_load
    b.y = Bs[(k + koff + 1) * 16 + l15];
    acc = __builtin_amdgcn_wmma_f32_16x16x4_f32(false, a, false, b,
                                                (short)0, acc, false, false);
  }
  const int row = tm + half * 8;
  const int col = tn + l15;
  float bv = bias ? bias[col] : 0.0f;
#pragma unroll
  for (int r = 0; r < 8; ++r)
    C[(size_t)(row + r) * Nc + col] = acc[r] + bv;
}

// O[16x16] += S^T (K x 16) @ X (K x 16) ; K split across waves, atomic combine.
__global__ void gat_wmma_stx(const float* __restrict__ S,
                             const float* __restrict__ X,
                             float* __restrict__ O, int Krows) {
  const int lane = threadIdx.x & 31;
  const int wave = (blockIdx.x * blockDim.x + threadIdx.x) >> 5;
  const int nw   = (gridDim.x * blockDim.x) >> 5;
  const int half = lane >> 4, l15 = lane & 15, koff = half * 2;
  v8f acc = {};
  for (int k = wave * 4; k < Krows; k += nw * 4) {   // uniform trip count per wave
    v2f a, b;
    a.x = S[(size_t)(k + koff) * CCL + l15];          // A[m][k] = S[k][m]
    a.y = S[(size_t)(k + koff + 1) * CCL + l15];
    b.x = X[(size_t)(k + koff) * CCL + l15];
    b.y = X[(size_t)(k + koff + 1) * CCL + l15];
    acc = __builtin_amdgcn_wmma_f32_16x16x4_f32(false, a, false, b,
                                                (short)0, acc, false, false);
  }
  const int row = half * 8, col = l15;
#pragma unroll
  for (int r = 0; r < 8; ++r)
    atomicAdd(&O[(row + r) * CCL + col], acc[r]);
}

// asrc[n] = g[n].att_src ; adst[n] = g[n].att_dst   (one wave per row)
__global__ void gat_row_dot2(const float* __restrict__ g,
                             const float* __restrict__ att_s,
                             const float* __restrict__ att_d,
                             float* __restrict__ asrc, float* __restrict__ adst,
                             int Nr, int dout) {
  const int lane = threadIdx.x & 31;
  int wave = (blockIdx.x * blockDim.x + threadIdx.x) >> 5;
  const int nw = (gridDim.x * blockDim.x) >> 5;
  for (int n = wave; n < Nr; n += nw) {
    float s1 = 0.f, s2 = 0.f;
    for (int c = lane; c < dout; c += 32) {
      float v = g[(size_t)n * dout + c];
      s1 += v * att_s[c];
      s2 += v * att_d[c];
    }
#pragma unroll
    for (int off = 16; off > 0; off >>= 1) {
      s1 += __shfl_xor(s1, off);
      s2 += __shfl_xor(s2, off);
    }
    if (lane == 0) { asrc[n] = s1; adst[n] = s2; }
  }
}

// pass A: leaky score per edge + per-destination running max (key-encoded)
__global__ void gat_edge_score(const float* __restrict__ asrc,
                               const float* __restrict__ adst,
                               const int* __restrict__ ei,
                               float* __restrict__ score,
                               unsigned* __restrict__ mkey) {
  const int total = EE + NN;
  for (int e = blockIdx.x * blockDim.x + threadIdx.x; e < total;
       e += gridDim.x * blockDim.x) {
    int s = (e < EE) ? ei[e] : (e - EE);
    int d = (e < EE) ? ei[EE + e] : (e - EE);
    float v = asrc[s] + adst[d];
    v = v > 0.f ? v : NEGS * v;                     // LeakyReLU
    score[e] = v;
    atomicMax(&mkey[d], f2key(v));
  }
}

// pass B: ex = exp(score - max[dst]) ; den[dst] += ex
__global__ void gat_edge_exp(const float* __restrict__ score,
                             const unsigned* __restrict__ mkey,
                             const int* __restrict__ ei,
                             float* __restrict__ ex, float* __restrict__ den) {
  const int total = EE + NN;
  for (int e = blockIdx.x * blockDim.x + threadIdx.x; e < total;
       e += gridDim.x * blockDim.x) {
    int d = (e < EE) ? ei[EE + e] : (e - EE);
    float t = __expf(score[e] - key2f(mkey[d]));
    ex[e] = t;
    atomicAdd(&den[d], t);
  }
}

// pass C: agg[dst] += (ex/den) * g[src]   (one wave per edge, L2-resident
// random gather -> prefetch next edge's source row to hide latency)
__global__ void gat_edge_agg(const float* __restrict__ g,
                             const float* __restrict__ ex,
                             const float* __restrict__ den,
                             const int* __restrict__ ei,
                             float* __restrict__ agg, int dout) {
  const int lane = threadIdx.x & 31;
  int wave = (blockIdx.x * blockDim.x + threadIdx.x) >> 5;
  const int nw = (gridDim.x * blockDim.x) >> 5;
  const int total = EE + NN;
  for (int e = wave; e < total; e += nw) {
    int eN = e + nw;                                 // prefetch next gather row
    if (eN < total) {
      int sn = (eN < EE) ? ei[eN] : (eN - EE);
      __builtin_prefetch(&g[(size_t)sn * dout + lane], 0, 0);
    }
    int s = (e < EE) ? ei[e] : (e - EE);
    int d = (e < EE) ? ei[EE + e] : (e - EE);
    float a = ex[e] / (den[d] + 1e-16f);
    for (int c = lane; c < dout; c += 32)
      atomicAdd(&agg[(size_t)d * dout + c], a * g[(size_t)s * dout + c]);
  }
}

// column sums / sumsq for batch-norm stats (blockDim.x == dout)
__global__ void gat_bn_stats(const float* __restrict__ v,
                             float* __restrict__ sum, float* __restrict__ sq,
                             int Nr, int dout, int rowsPerBlk) {
  const int c = threadIdx.x;
  int r0 = blockIdx.x * rowsPerBlk;
  int r1 = r0 + rowsPerBlk; if (r1 > Nr) r1 = Nr;
  float s = 0.f, q = 0.f;
  for (int r = r0; r < r1; ++r) {
    float x = v[(size_t)r * dout + c];
    s += x; q += x * x;
  }
  atomicAdd(&sum[c], s);
  atomicAdd(&sq[c], q);
}

// h = relu(gamma*(v-mu)*rsqrt(var+eps)+beta)   (bias cancels under mean-centering)
__global__ void gat_bn_apply_relu(const float* __restrict__ v,
                                  const float* __restrict__ sum,
                                  const float* __restrict__ sq,
                                  const float* __restrict__ gamma,
                                  const float* __restrict__ beta,
                                  float* __restrict__ out, int Nr, int dout) {
  const size_t total = (size_t)Nr * dout;
  for (size_t i = blockIdx.x * (size_t)blockDim.x + threadIdx.x; i < total;
       i += (size_t)gridDim.x * blockDim.x) {
    int c = (int)(i % dout);
    float mu  = sum[c] / Nr;
    float var = sq[c] / Nr - mu * mu;
    float x = gamma[c] * (v[i] - mu) * rsqrtf(var + BNEPS) + beta[c];
    out[i] = x > 0.f ? x : 0.f;
  }
}

// row softmax over C=16 -> s, plus column sums for the cluster loss
__global__ void gat_softmax16(const float* __restrict__ pre,
                              float* __restrict__ sout,
                              float* __restrict__ colsum) {
  int n = blockIdx.x * blockDim.x + threadIdx.x;
  if (n >= NN) return;
  float v[CCL]; float mx = -3.0e38f;
#pragma unroll
  for (int c = 0; c < CCL; ++c) { v[c] = pre[(size_t)n * CCL + c]; mx = fmaxf(mx, v[c]); }
  float s = 0.f;
#pragma unroll
  for (int c = 0; c < CCL; ++c) { v[c] = __expf(v[c] - mx); s += v[c]; }
  float inv = 1.f / s;
#pragma unroll
  for (int c = 0; c < CCL; ++c) {
    float t = v[c] * inv;
    sout[(size_t)n * CCL + c] = t;
    atomicAdd(&colsum[c], t);
  }
}

// t[src] += s[dst] per original edge (t = row-accumulated adj@s);  deg[src] += 1
__global__ void gat_scatter_adj(const float* __restrict__ s,
                                const int* __restrict__ ei,
                                float* __restrict__ t, float* __restrict__ deg) {
  for (int e = blockIdx.x * blockDim.x + threadIdx.x; e < EE;
       e += gridDim.x * blockDim.x) {
    int si = ei[e], di = ei[EE + e];
    atomicAdd(&deg[si], 1.0f);
#pragma unroll
    for (int c = 0; c < CCL; ++c)
      atomicAdd(&t[(size_t)si * CCL + c], s[(size_t)di * CCL + c]);
  }
}

// ca[c] = sum_n s[n][c] * deg[n]
__global__ void gat_ca_reduce(const float* __restrict__ s,
                              const float* __restrict__ deg,
                              float* __restrict__ ca) {
  float acc[CCL];
#pragma unroll
  for (int c = 0; c < CCL; ++c) acc[c] = 0.f;
  for (int n = blockIdx.x * blockDim.x + threadIdx.x; n < NN;
       n += gridDim.x * blockDim.x) {
    float dg = deg[n];
#pragma unroll
    for (int c = 0; c < CCL; ++c) acc[c] += s[(size_t)n * CCL + c] * dg;
  }
#pragma unroll
  for (int c = 0; c < CCL; ++c) atomicAdd(&ca[c], acc[c]);
}

// spectral + ortho + cluster -> single scalar
__global__ void gat_dmon_loss(const float* __restrict__ oadj,
                              const float* __restrict__ ss,
                              const float* __restrict__ ca,
                              const float* __restrict__ colsum,
                              float m, float* __restrict__ loss) {
  if (threadIdx.x != 0 || blockIdx.x != 0) return;
  float tr = 0.f, ca2 = 0.f, fn = 0.f, cs = 0.f;
  for (int c = 0; c < CCL; ++c) { tr += oadj[c * CCL + c]; ca2 += ca[c] * ca[c]; cs += colsum[c] * colsum[c]; }
  float spectral = -(tr - ca2 / (2.f * m)) / (2.f * m);
  for (int i = 0; i < CCL * CCL; ++i) fn += ss[i] * ss[i];
  fn = sqrtf(fn);
  float on = 0.f;
  for (int i = 0; i < CCL * CCL; ++i) {
    float v = ss[i] / fn - ((i % (CCL + 1) == 0) ? 0.25f : 0.f);   // I/sqrt(16)
    on += v * v;
  }
  float ortho = sqrtf(on);
  float cluster = sqrtf(cs) / (float)NN * 4.f - 1.f;               // sqrt(C)=4
  loss[0] = spectral + ortho + cluster;
}

// z = (v + bias) / max(||v+bias||, 1e-12)
__global__ void gat_l2norm16(const float* __restrict__ v,
                             const float* __restrict__ bias,
                             float* __restrict__ z) {
  int n = blockIdx.x * blockDim.x + threadIdx.x;
  if (n >= NN) return;
  float w[CCL]; float s = 0.f;
#pragma unroll
  for (int c = 0; c < CCL; ++c) { w[c] = v[(size_t)n * CCL + c] + bias[c]; s += w[c] * w[c]; }
  float inv = 1.f / fmaxf(sqrtf(s), 1e-12f);
#pragma unroll
  for (int c = 0; c < CCL; ++c) z[(size_t)n * CCL + c] = w[c] * inv;
}

// ---------------------------------------------------------------------
extern "C" void kernel_launch(void* const* d_in, const int* in_sizes, int n_in,
                              void* d_out, int out_size, void* d_ws, size_t ws_size,
                              hipStream_t stream) {
  (void)in_sizes; (void)n_in; (void)out_size; (void)ws_size;

  // --- input pytree flattening (sorted-key order inside params) ---
  const float* x  = (const float*)d_in[0];
  // d_in[1] = edge_weight: ignored by GATConv(edge_dim=None)
  const float* bn_beta[5], *bn_gamma[5];
  for (int l = 0; l < 5; ++l) { bn_beta[l] = (const float*)d_in[2 + 2 * l];
                                bn_gamma[l] = (const float*)d_in[3 + 2 * l]; }
  const float* cW[5], *cAd[5], *cAs[5], *cB[5];
  for (int l = 0; l < 5; ++l) {
    int b = 12 + 4 * l;
    cW[l]  = (const float*)d_in[b + 0];   // W
    cAd[l] = (const float*)d_in[b + 1];   // att_dst
    cAs[l] = (const float*)d_in[b + 2];   // att_src
    cB[l]  = (const float*)d_in[b + 3];   // bias
  }
  const float* W1 = (const float*)d_in[32];
  const float* W2 = (const float*)d_in[33];
  const float* b1 = (const float*)d_in[34];
  const float* b2 = (const float*)d_in[35];
  const int*   ei = (const int*)d_in[36];

  float* s_out = (float*)d_out;                // [NN x 16]
  float* z_out = s_out + (size_t)NN * CCL;     // [NN x 16]
  float* l_out = z_out + (size_t)NN * CCL;     // [1]

  // --- workspace carve-up (fp32 units, 256-elem aligned) ---
  float* ws = (float*)d_ws;
  size_t off = 0;
  auto carve = [&](size_t n) { size_t r = off; off += (n + 255) & ~(size_t)255; return ws + r; };
  float*    g     = carve((size_t)NN * HH);
  float*    hbuf  = carve((size_t)NN * HH);
  float*    agg   = carve((size_t)NN * HH);
  float*    asrc  = carve(NN);
  float*    adst  = carve(NN);
  unsigned* mkey  = (unsigned*)carve(NN);
  float*    den   = carve(NN);
  float*    score = carve(EE + NN);
  float*    exb   = carve(EE + NN);
  float*    bnsum = carve(HH);
  float*    bnsq  = carve(HH);
  float*    tbuf  = carve((size_t)NN * CCL);
  float*    deg   = carve(NN);
  float*    g16   = carve((size_t)NN * CCL);
  float*    agg16 = carve((size_t)NN * CCL);
  float*    oadj  = carve(256);
  float*    ssb   = carve(256);
  float*    cab   = carve(16);
  float*    csum  = carve(16);

  const int ETOT = EE + NN;
  dim3 b256(256);
  dim3 gEdges((ETOT + 255) / 256);

  // ---------------- 4x GATConv + BN + ReLU ----------------
  const float* hin = x; int K = FIN;
  for (int l = 0; l < 4; ++l) {
    // g = hin @ W[l]  (WMMA; 4 waves/block share LDS B panel)
    gat_wmma_gemm_f32<<<dim3(NN / 64, HH / 16), dim3(128), 0, stream>>>(
        hin, cW[l], nullptr, g, NN, K, HH);
    gat_row_dot2<<<dim3(256), b256, 0, stream>>>(g, cAs[l], cAd[l], asrc, adst, NN, HH);

    hipMemsetAsync(mkey, 0, NN * sizeof(unsigned), stream);        // key 0 == most-negative
    hipMemsetAsync(den,  0, NN * sizeof(float), stream);
    hipMemsetAsync(agg,  0, (size_t)NN * HH * sizeof(float), stream);

    gat_edge_score<<<gEdges, b256, 0, stream>>>(asrc, adst, ei, score, mkey);
    gat_edge_exp  <<<gEdges, b256, 0, stream>>>(score, mkey, ei, exb, den);
    gat_edge_agg  <<<dim3(ETOT / 8), b256, 0, stream>>>(g, exb, den, ei, agg, HH);

    hipMemsetAsync(bnsum, 0, HH * sizeof(float), stream);
    hipMemsetAsync(bnsq,  0, HH * sizeof(float), stream);
    gat_bn_stats<<<dim3(NN / 64), dim3(HH), 0, stream>>>(agg, bnsum, bnsq, NN, HH, 64);
    gat_bn_apply_relu<<<dim3(2048), b256, 0, stream>>>(
        agg, bnsum, bnsq, bn_gamma[l], bn_beta[l], hbuf, NN, HH);

    hin = hbuf; K = HH;
  }

  // ---------------- DMoN pooling ----------------
  // g = hbuf @ W1 + b1 ; g16 = g @ W2 + b2   (WMMA)
  gat_wmma_gemm_f32<<<dim3(NN / 64, HH / 16), dim3(128), 0, stream>>>(
      hbuf, W1, b1, g, NN, HH, HH);
  gat_wmma_gemm_f32<<<dim3(NN / 64, CCL / 16), dim3(128), 0, stream>>>(
      g, W2, b2, g16, NN, HH, CCL);

  hipMemsetAsync(csum, 0, 16 * sizeof(float), stream);
  gat_softmax16<<<dim3(NN / 256), b256, 0, stream>>>(g16, s_out, csum);

  hipMemsetAsync(tbuf, 0, (size_t)NN * CCL * sizeof(float), stream);
  hipMemsetAsync(deg,  0, NN * sizeof(float), stream);
  gat_scatter_adj<<<dim3(EE / 256), b256, 0, stream>>>(s_out, ei, tbuf, deg);

  hipMemsetAsync(oadj, 0, 256 * sizeof(float), stream);
  hipMemsetAsync(ssb,  0, 256 * sizeof(float), stream);
  gat_wmma_stx<<<dim3(32), dim3(128), 0, stream>>>(s_out, tbuf,  oadj, NN);  // s^T (adj s)
  gat_wmma_stx<<<dim3(32), dim3(128), 0, stream>>>(s_out, s_out, ssb,  NN);  // s^T s

  hipMemsetAsync(cab, 0, 16 * sizeof(float), stream);
  gat_ca_reduce<<<dim3(32), b256, 0, stream>>>(s_out, deg, cab);
  gat_dmon_loss<<<dim3(1), dim3(32), 0, stream>>>(oadj, ssb, cab, csum,
                                                  (float)EE, l_out);

  // ---------------- final GATConv (H -> 16) + L2 normalize ----------------
  gat_wmma_gemm_f32<<<dim3(NN / 64, CCL / 16), dim3(128), 0, stream>>>(
      hbuf, cW[4], nullptr, g16, NN, HH, CCL);
  gat_row_dot2<<<dim3(256), b256, 0, stream>>>(g16, cAs[4], cAd[4], asrc, adst, NN, CCL);
  hipMemsetAsync(mkey,  0, NN * sizeof(unsigned), stream);
  hipMemsetAsync(den,   0, NN * sizeof(float), stream);
  hipMemsetAsync(agg16, 0, (size_t)NN * CCL * sizeof(float), stream);
  gat_edge_score<<<gEdges, b256, 0, stream>>>(asrc, adst, ei, score, mkey);
  gat_edge_exp  <<<gEdges, b256, 0, stream>>>(score, mkey, ei, exb, den);
  gat_edge_agg  <<<dim3(ETOT / 8), b256, 0, stream>>>(g16, exb, den, ei, agg16, CCL);
  gat_l2norm16<<<dim3(NN / 256), b256, 0, stream>>>(agg16, cB[4], z_out);
}